// HybridNATModel_65481071402235
// MI455X (gfx1250) — compile-verified
//
#include <hip/hip_runtime.h>
#include <hip/hip_bf16.h>
#include <math.h>

// ---------------------------------------------------------------------------
// HybridNAT pipeline for MI455X (gfx1250, wave32).
//   B=512, C_IN=1, HW=28, S=2048, D=128, NC=10
// ---------------------------------------------------------------------------

typedef float v2f __attribute__((ext_vector_type(2)));
typedef float v4f __attribute__((ext_vector_type(4)));
typedef float v8f __attribute__((ext_vector_type(8)));
typedef int   v4i __attribute__((ext_vector_type(4)));

#define NB   512
#define NS   2048
#define ND   128
#define NCLS 10

// Async global->LDS path (gfx1250 GLOBAL_LOAD_ASYNC_TO_LDS_B128, ASYNCcnt).
// Builtin signature (from clang diagnostics): 
//   void(v4i addrspace(1)*, v4i addrspace(3)*, imm int offset, imm int cpol)
#if defined(__has_builtin)
#if __has_builtin(__builtin_amdgcn_global_load_async_to_lds_b128) && \
    __has_builtin(__builtin_amdgcn_s_wait_asynccnt)
#define USE_ASYNC_LDS 1
#endif
#endif

#ifdef USE_ASYNC_LDS
typedef __attribute__((address_space(1))) v4i* gv4i_t;
typedef __attribute__((address_space(3))) v4i* lv4i_t;
#endif

// ---------------------------------------------------------------------------
// Kernel 1: fused conv1+relu+pool -> conv2+relu+pool per image.
// One block (256 threads = 8 waves) per image; everything staged in LDS.
// ---------------------------------------------------------------------------
__global__ __launch_bounds__(256) void conv_feat_kernel(
    const float* __restrict__ x,       // [B,1,28,28]
    const float* __restrict__ w1,      // [8,1,3,3]
    const float* __restrict__ b1,      // [8]
    const float* __restrict__ w2,      // [16,8,3,3]
    const float* __restrict__ b2,      // [16]
    float* __restrict__ h_flat)        // [B,784]
{
    __shared__ float s_in[28 * 28];
    __shared__ float s_p1[8 * 14 * 14];
    __shared__ float s_w1[8 * 9];
    __shared__ float s_b1[8];
    __shared__ float s_w2[16 * 8 * 9];
    __shared__ float s_b2[16];

    const int tid = threadIdx.x;
    const int img = blockIdx.x;
    const float* xin = x + img * 784;

    for (int i = tid; i < 784; i += 256)  s_in[i] = xin[i];
    for (int i = tid; i < 72; i += 256)   s_w1[i] = w1[i];
    for (int i = tid; i < 1152; i += 256) s_w2[i] = w2[i];
    if (tid < 8)  s_b1[tid] = b1[tid];
    if (tid < 16) s_b2[tid] = b2[tid];
    __syncthreads();

    // conv1 (SAME, 3x3, 1->8) + relu + 2x2 maxpool -> s_p1[8][14][14]
    for (int i = tid; i < 8 * 14 * 14; i += 256) {
        const int oc  = i / 196;
        const int rem = i % 196;
        const int py  = rem / 14;
        const int px  = rem % 14;
        float m = 0.0f;  // relu output >= 0
        #pragma unroll
        for (int dy = 0; dy < 2; ++dy) {
            #pragma unroll
            for (int dx = 0; dx < 2; ++dx) {
                const int y = py * 2 + dy, xx = px * 2 + dx;
                float acc = s_b1[oc];
                #pragma unroll
                for (int ky = 0; ky < 3; ++ky) {
                    const int iy = y + ky - 1;
                    if (iy < 0 || iy >= 28) continue;
                    #pragma unroll
                    for (int kx = 0; kx < 3; ++kx) {
                        const int ix = xx + kx - 1;
                        if (ix < 0 || ix >= 28) continue;
                        acc = fmaf(s_in[iy * 28 + ix], s_w1[oc * 9 + ky * 3 + kx], acc);
                    }
                }
                m = fmaxf(m, fmaxf(acc, 0.0f));
            }
        }
        s_p1[i] = m;
    }
    __syncthreads();

    // conv2 (SAME, 3x3, 8->16) + relu + 2x2 maxpool -> h_flat[img][16*7*7]
    float* ho = h_flat + img * 784;
    for (int i = tid; i < 16 * 7 * 7; i += 256) {
        const int oc  = i / 49;
        const int rem = i % 49;
        const int py  = rem / 7;
        const int px  = rem % 7;
        float m = 0.0f;
        #pragma unroll
        for (int dy = 0; dy < 2; ++dy) {
            #pragma unroll
            for (int dx = 0; dx < 2; ++dx) {
                const int y = py * 2 + dy, xx = px * 2 + dx;
                float acc = s_b2[oc];
                for (int ic = 0; ic < 8; ++ic) {
                    const float* pin = s_p1 + ic * 196;
                    const float* pw  = s_w2 + oc * 72 + ic * 9;
                    #pragma unroll
                    for (int ky = 0; ky < 3; ++ky) {
                        const int iy = y + ky - 1;
                        if (iy < 0 || iy >= 14) continue;
                        #pragma unroll
                        for (int kx = 0; kx < 3; ++kx) {
                            const int ix = xx + kx - 1;
                            if (ix < 0 || ix >= 14) continue;
                            acc = fmaf(pin[iy * 14 + ix], pw[ky * 3 + kx], acc);
                        }
                    }
                }
                m = fmaxf(m, fmaxf(acc, 0.0f));
            }
        }
        ho[i] = m;
    }
}

// ---------------------------------------------------------------------------
// Kernel 2: FC GEMM via V_WMMA_F32_16X16X4_F32 with fused relu + sincos
// epilogue. Two interleaved accumulators to break the XDL dependency chain.
// One 16x16 tile per wave; M=512 -> 32 tiles, N=128 -> 8 tiles = 256 waves.
// ---------------------------------------------------------------------------
__global__ __launch_bounds__(256) void fc_wmma_kernel(
    const float* __restrict__ A,     // [512,784]
    const float* __restrict__ W,     // [128,784]
    const float* __restrict__ bias,  // [128]
    float* __restrict__ cf,          // [512,128]
    float* __restrict__ sf)          // [512,128]
{
    const int tid   = threadIdx.x;
    const int lane  = tid & 31;
    const int wave  = blockIdx.x * (blockDim.x >> 5) + (tid >> 5);
    const int tileM = wave >> 3;   // 0..31
    const int tileN = wave & 7;    // 0..7
    const int r16   = lane & 15;
    const int kb    = (lane >> 4) << 1;   // 0 or 2

    const float* arow = A + (size_t)(tileM * 16 + r16) * 784 + kb;
    const float* brow = W + (size_t)(tileN * 16 + r16) * 784 + kb;

    v8f acc0 = {}, acc1 = {};
    for (int k = 0; k < 784; k += 8) {
        v2f a0 = *(const v2f*)(arow + k);
        v2f b0 = *(const v2f*)(brow + k);
        acc0 = __builtin_amdgcn_wmma_f32_16x16x4_f32(false, a0, false, b0,
                                                     (short)0, acc0, false, false);
        v2f a1 = *(const v2f*)(arow + k + 4);
        v2f b1 = *(const v2f*)(brow + k + 4);
        acc1 = __builtin_amdgcn_wmma_f32_16x16x4_f32(false, a1, false, b1,
                                                     (short)0, acc1, false, false);
    }
    v8f acc = acc0 + acc1;

    // C/D layout: VGPR r -> row tileM*16 + r + 8*(lane>>4), col = lane&15.
    const int col     = tileN * 16 + r16;
    const int rowbase = tileM * 16 + ((lane >> 4) << 3);
    const float bv    = bias[col];
    #pragma unroll
    for (int r = 0; r < 8; ++r) {
        float f = fmaxf(acc[r] + bv, 0.0f);
        float s, c;
        __sincosf(0.5f * f, &s, &c);
        const int idx = (rowbase + r) * ND + col;
        cf[idx] = c;
        sf[idx] = s;
    }
}

// ---------------------------------------------------------------------------
// Kernel 3: support trig precompute. cs = cos(support/2), ss = sin(support/2).
// ---------------------------------------------------------------------------
__global__ __launch_bounds__(256) void support_trig_kernel(
    const float* __restrict__ support,  // [S*D]
    float* __restrict__ cs,
    float* __restrict__ ss)
{
    const int i = blockIdx.x * 256 + threadIdx.x;
    if (i < NS * ND) {
        float s, c;
        __sincosf(0.5f * support[i], &s, &c);
        cs[i] = c;
        ss[i] = s;
    }
}

// ---------------------------------------------------------------------------
// Kernel 3b: zero-pad classifier weights [10,2048] -> [16,2048].
// ---------------------------------------------------------------------------
__global__ __launch_bounds__(256) void pad_cls_kernel(
    const float* __restrict__ W,   // [10,2048]
    float* __restrict__ Wp)        // [16,2048]
{
    const int i = blockIdx.x * 256 + threadIdx.x;
    if (i < 16 * NS) {
        const int row = i / NS;
        Wp[i] = (row < NCLS) ? W[i] : 0.0f;
    }
}

// ---------------------------------------------------------------------------
// Kernel 4: fidelity kernel matrix.
//   kv[b][s] = prod_d ( cf[b][d]*cs[s][d] + sf[b][d]*ss[s][d] )^2
// 32x32 output tile per block; 64 KB LDS. s-indexed arrays use an XOR float4
// chunk swizzle (chunk c of row r stored at c^r) so a wave32 ds_load_b128
// spreads 128 dwords evenly over the 64 banks (2-cycle optimum, no padding).
// b-indexed arrays are read wave-uniformly (broadcast) and stay linear.
// Staging uses GLOBAL_LOAD_ASYNC_TO_LDS_B128 (ASYNCcnt) when available.
// ---------------------------------------------------------------------------
__global__ __launch_bounds__(256) void fidelity_kernel(
    const float* __restrict__ cf, const float* __restrict__ sf,   // [B,D]
    const float* __restrict__ cs, const float* __restrict__ ss,   // [S,D]
    float* __restrict__ kv)                                       // [B,S]
{
    __shared__ float s_cf[32 * ND];
    __shared__ float s_sf[32 * ND];
    __shared__ float s_cs[32 * ND];   // XOR-swizzled float4 chunks
    __shared__ float s_ss[32 * ND];   // XOR-swizzled float4 chunks

    const int tid = threadIdx.x;
    const int b0  = blockIdx.y * 32;
    const int s0  = blockIdx.x * 32;

    // Stage tiles: 32 rows x 32 float4-chunks per array, 4 chunks/thread/array.
#ifdef USE_ASYNC_LDS
    for (int i = tid; i < 32 * 32; i += 256) {
        const int r  = i >> 5;
        const int c  = i & 31;
        const int dc = c << 2;
        const int pc = ((c ^ r) & 31) << 2;   // swizzled chunk position
        __builtin_amdgcn_global_load_async_to_lds_b128(
            (gv4i_t)(cf + (size_t)(b0 + r) * ND + dc), (lv4i_t)(s_cf + r * ND + dc), 0, 0);
        __builtin_amdgcn_global_load_async_to_lds_b128(
            (gv4i_t)(sf + (size_t)(b0 + r) * ND + dc), (lv4i_t)(s_sf + r * ND + dc), 0, 0);
        __builtin_amdgcn_global_load_async_to_lds_b128(
            (gv4i_t)(cs + (size_t)(s0 + r) * ND + dc), (lv4i_t)(s_cs + r * ND + pc), 0, 0);
        __builtin_amdgcn_global_load_async_to_lds_b128(
            (gv4i_t)(ss + (size_t)(s0 + r) * ND + dc), (lv4i_t)(s_ss + r * ND + pc), 0, 0);
    }
    __builtin_amdgcn_s_wait_asynccnt(0);
    __syncthreads();
#else
    for (int i = tid; i < 32 * 32; i += 256) {
        const int r  = i >> 5;
        const int c  = i & 31;
        const int dc = c << 2;
        const int pc = ((c ^ r) & 31) << 2;
        *(v4f*)(s_cf + r * ND + dc) = *(const v4f*)(cf + (size_t)(b0 + r) * ND + dc);
        *(v4f*)(s_sf + r * ND + dc) = *(const v4f*)(sf + (size_t)(b0 + r) * ND + dc);
        *(v4f*)(s_cs + r * ND + pc) = *(const v4f*)(cs + (size_t)(s0 + r) * ND + dc);
        *(v4f*)(s_ss + r * ND + pc) = *(const v4f*)(ss + (size_t)(s0 + r) * ND + dc);
    }
    __syncthreads();
#endif

    const int sl = tid & 31;   // s within tile (per-lane -> swizzled reads)
    const int bg = tid >> 5;   // 0..7 (wave-uniform -> broadcast reads)

    float p0 = 1.0f, p1 = 1.0f, p2 = 1.0f, p3 = 1.0f;
    #pragma unroll 2
    for (int c = 0; c < 32; ++c) {
        const int pc = ((c ^ sl) & 31) << 2;  // undo swizzle for this lane's row
        const int dc = c << 2;
        const v4f cv = *(const v4f*)(s_cs + sl * ND + pc);
        const v4f sv = *(const v4f*)(s_ss + sl * ND + pc);

        const v4f c0 = *(const v4f*)(s_cf + (bg     ) * ND + dc);
        const v4f f0 = *(const v4f*)(s_sf + (bg     ) * ND + dc);
        const v4f c1 = *(const v4f*)(s_cf + (bg +  8) * ND + dc);
        const v4f f1 = *(const v4f*)(s_sf + (bg +  8) * ND + dc);
        const v4f c2 = *(const v4f*)(s_cf + (bg + 16) * ND + dc);
        const v4f f2 = *(const v4f*)(s_sf + (bg + 16) * ND + dc);
        const v4f c3 = *(const v4f*)(s_cf + (bg + 24) * ND + dc);
        const v4f f3 = *(const v4f*)(s_sf + (bg + 24) * ND + dc);

        const v4f t0 = c0 * cv + f0 * sv;   // contracts to v_fma
        const v4f t1 = c1 * cv + f1 * sv;
        const v4f t2 = c2 * cv + f2 * sv;
        const v4f t3 = c3 * cv + f3 * sv;

        const float q0 = (t0.x * t0.y) * (t0.z * t0.w);
        const float q1 = (t1.x * t1.y) * (t1.z * t1.w);
        const float q2 = (t2.x * t2.y) * (t2.z * t2.w);
        const float q3 = (t3.x * t3.y) * (t3.z * t3.w);
        p0 *= q0 * q0;
        p1 *= q1 * q1;
        p2 *= q2 * q2;
        p3 *= q3 * q3;
    }

    const int sc = s0 + sl;
    kv[(size_t)(b0 + bg)      * NS + sc] = p0;
    kv[(size_t)(b0 + bg + 8)  * NS + sc] = p1;
    kv[(size_t)(b0 + bg + 16) * NS + sc] = p2;
    kv[(size_t)(b0 + bg + 24) * NS + sc] = p3;
}

// ---------------------------------------------------------------------------
// Kernel 5: classifier GEMM via V_WMMA_F32_16X16X4_F32 on zero-padded W.
//   logits[b][c] = sum_s kv[b][s] * Wp[c][s] + cls_b[c]
// M=512 -> 32 waves; N=16 (cols 10..15 are zero rows of Wp).
// ---------------------------------------------------------------------------
__global__ __launch_bounds__(256) void cls_wmma_kernel(
    const float* __restrict__ kv,     // [512,2048]
    const float* __restrict__ Wp,     // [16,2048] zero-padded
    const float* __restrict__ bias,   // [10]
    float* __restrict__ out)          // [512,10]
{
    const int tid   = threadIdx.x;
    const int lane  = tid & 31;
    const int tileM = blockIdx.x * (blockDim.x >> 5) + (tid >> 5);  // 0..31
    const int r16   = lane & 15;
    const int kb    = (lane >> 4) << 1;

    const float* arow = kv + (size_t)(tileM * 16 + r16) * NS + kb;
    const float* brow = Wp + (size_t)r16 * NS + kb;

    v8f acc0 = {}, acc1 = {};
    for (int k = 0; k < NS; k += 8) {
        v2f a0 = *(const v2f*)(arow + k);
        v2f b0 = *(const v2f*)(brow + k);
        acc0 = __builtin_amdgcn_wmma_f32_16x16x4_f32(false, a0, false, b0,
                                                     (short)0, acc0, false, false);
        v2f a1 = *(const v2f*)(arow + k + 4);
        v2f b1 = *(const v2f*)(brow + k + 4);
        acc1 = __builtin_amdgcn_wmma_f32_16x16x4_f32(false, a1, false, b1,
                                                     (short)0, acc1, false, false);
    }
    v8f acc = acc0 + acc1;

    const int col     = r16;
    const int rowbase = tileM * 16 + ((lane >> 4) << 3);
    if (col < NCLS) {
        const float bv = bias[col];
        #pragma unroll
        for (int r = 0; r < 8; ++r)
            out[(rowbase + r) * NCLS + col] = acc[r] + bv;
    }
}

// ---------------------------------------------------------------------------
// Launch
// ---------------------------------------------------------------------------
extern "C" void kernel_launch(void* const* d_in, const int* in_sizes, int n_in,
                              void* d_out, int out_size, void* d_ws, size_t ws_size,
                              hipStream_t stream) {
    (void)in_sizes; (void)n_in; (void)out_size; (void)ws_size;

    const float* x       = (const float*)d_in[0];
    const float* conv1_w = (const float*)d_in[1];
    const float* conv1_b = (const float*)d_in[2];
    const float* conv2_w = (const float*)d_in[3];
    const float* conv2_b = (const float*)d_in[4];
    const float* fc_w    = (const float*)d_in[5];
    const float* fc_b    = (const float*)d_in[6];
    const float* support = (const float*)d_in[7];
    const float* cls_w   = (const float*)d_in[8];
    const float* cls_b   = (const float*)d_in[9];
    float* logits        = (float*)d_out;

    // Workspace layout (floats)
    float* ws     = (float*)d_ws;
    float* h_flat = ws;                       // 512*784
    float* cf     = h_flat + NB * 784;        // 512*128
    float* sf     = cf + NB * ND;             // 512*128
    float* cs     = sf + NB * ND;             // 2048*128
    float* ss     = cs + NS * ND;             // 2048*128
    float* kv     = ss + NS * ND;             // 512*2048
    float* wpad   = kv + (size_t)NB * NS;     // 16*2048

    // 1) conv stack: one image per block
    conv_feat_kernel<<<NB, 256, 0, stream>>>(x, conv1_w, conv1_b,
                                             conv2_w, conv2_b, h_flat);

    // 2) support trig + classifier weight padding (independent of conv)
    support_trig_kernel<<<(NS * ND + 255) / 256, 256, 0, stream>>>(support, cs, ss);
    pad_cls_kernel<<<(16 * NS + 255) / 256, 256, 0, stream>>>(cls_w, wpad);

    // 3) FC GEMM (WMMA f32) + relu + sincos epilogue: 256 tiles = 32 blocks x 8 waves
    fc_wmma_kernel<<<32, 256, 0, stream>>>(h_flat, fc_w, fc_b, cf, sf);

    // 4) fidelity kernel matrix: 64 x 16 blocks of 32x32 tiles
    fidelity_kernel<<<dim3(NS / 32, NB / 32), 256, 0, stream>>>(cf, sf, cs, ss, kv);

    // 5) classifier GEMM (WMMA f32): 32 waves = 4 blocks x 8 waves
    cls_wmma_kernel<<<4, 256, 0, stream>>>(kv, wpad, cls_b, logits);
}